// GNNModel_33595234189756
// MI455X (gfx1250) — compile-verified
//
#include <hip/hip_runtime.h>

typedef __attribute__((ext_vector_type(2))) float v2f;
typedef __attribute__((ext_vector_type(8))) float v8f;

#define D_IN  128
#define D_HID 128
#define D_OUT 64

// ---------------------------------------------------------------------------
// Degree / normalization kernels
// ---------------------------------------------------------------------------
__global__ void deg_init_kernel(float* __restrict__ deg, int n) {
    int i = blockIdx.x * blockDim.x + threadIdx.x;
    if (i < n) deg[i] = 1.0f;  // self loop contributes 1
}

__global__ void deg_accum_kernel(const int* __restrict__ dst,
                                 float* __restrict__ deg, int e) {
    int i = blockIdx.x * blockDim.x + threadIdx.x;
    if (i < e) atomicAdd(&deg[dst[i]], 1.0f);
}

__global__ void deg_rsqrt_kernel(float* __restrict__ dinv, int n) {
    int i = blockIdx.x * blockDim.x + threadIdx.x;
    if (i < n) {
        float d = dinv[i];
        dinv[i] = (d > 0.0f) ? rsqrtf(d) : 0.0f;
    }
}

// ---------------------------------------------------------------------------
// f32 WMMA GEMM: H[M,NOUT] = A[M,128] @ W[128,NOUT]
// One block = 16-row strip of A staged in LDS; one wave per 16x16 output tile.
// Uses V_WMMA_F32_16X16X4_F32 (full fp32 precision, K stepped by 4).
// A 16x4 layout: lanes 0-15 -> M, {v0,v1} = K {k0,k0+1}; lanes 16-31 -> K {k0+2,k0+3}.
// B 4x16 layout mirrored; C/D: VGPR v -> row v + 8*(lane>=16), col = lane%16.
// ---------------------------------------------------------------------------
template <int NOUT>
__global__ void gemm_wmma_f32_kernel(const float* __restrict__ A,
                                     const float* __restrict__ W,
                                     float* __restrict__ H, int M) {
    constexpr int K = 128;
    constexpr int NT = NOUT / 16;          // waves per block
    __shared__ __align__(16) float lds_a[16 * K];   // 8 KB strip

    const int block_row = blockIdx.x * 16;
    const int tid = threadIdx.x;           // blockDim.x == 32*NT
    constexpr int NTHREADS = 32 * NT;

    // cooperative strip load, float4 per iteration
    for (int i = tid * 4; i < 16 * K; i += NTHREADS * 4) {
        *(float4*)(&lds_a[i]) = *(const float4*)(&A[(size_t)block_row * K + i]);
    }
    __syncthreads();

    const int wave = tid >> 5;
    const int lane = tid & 31;
    const int half = lane >> 4;            // 0: K k0,k0+1 | 1: K k0+2,k0+3
    const int l16  = lane & 15;
    const int col_base = wave * 16;

    v8f acc = {};
    #pragma unroll 4
    for (int k0 = 0; k0 < K; k0 += 4) {
        const int ka = k0 + half * 2;
        v2f a, b;
        a.x = lds_a[l16 * K + ka];
        a.y = lds_a[l16 * K + ka + 1];
        b.x = W[(size_t)ka * NOUT + col_base + l16];
        b.y = W[(size_t)(ka + 1) * NOUT + col_base + l16];
        acc = __builtin_amdgcn_wmma_f32_16x16x4_f32(
            /*neg_a=*/false, a, /*neg_b=*/false, b,
            /*c_mod=*/(short)0, acc, /*reuse_a=*/false, /*reuse_b=*/false);
    }

    #pragma unroll
    for (int v = 0; v < 8; ++v) {
        const int row = v + half * 8;
        H[(size_t)(block_row + row) * NOUT + col_base + l16] = acc[v];
    }
}

// ---------------------------------------------------------------------------
// agg[i][:] = h[i][:] * dinv[i]^2   (self-loop term; also serves as init)
// ---------------------------------------------------------------------------
template <int D>
__global__ void self_init_kernel(const float* __restrict__ h,
                                 const float* __restrict__ dinv,
                                 float* __restrict__ agg, int total) {
    int idx = blockIdx.x * blockDim.x + threadIdx.x;
    if (idx >= total) return;
    const int node = idx / D;
    const float di = dinv[node];
    agg[idx] = h[idx] * di * di;
}

// ---------------------------------------------------------------------------
// Edge scatter: one wave32 per edge; each lane handles VPL contiguous floats.
// agg[dst] += h[src] * dinv[src] * dinv[dst]  (f32 atomics, L2 resident)
// ---------------------------------------------------------------------------
template <int D, int VPL>
__global__ void edge_scatter_kernel(const int* __restrict__ src,
                                    const int* __restrict__ dst,
                                    const float* __restrict__ h,
                                    const float* __restrict__ dinv,
                                    float* __restrict__ agg, int e) {
    const int widx = (blockIdx.x * blockDim.x + threadIdx.x) >> 5;
    const int lane = threadIdx.x & 31;
    if (widx >= e) return;
    const int s = src[widx];
    const int d = dst[widx];
    const float norm = dinv[s] * dinv[d];
    const float* __restrict__ hp = h + (size_t)s * D + lane * VPL;
    float* __restrict__ op = agg + (size_t)d * D + lane * VPL;
    if constexpr (VPL == 4) {
        const float4 v = *(const float4*)hp;
        atomicAdd(op + 0, v.x * norm);
        atomicAdd(op + 1, v.y * norm);
        atomicAdd(op + 2, v.z * norm);
        atomicAdd(op + 3, v.w * norm);
    } else {
        const float2 v = *(const float2*)hp;
        atomicAdd(op + 0, v.x * norm);
        atomicAdd(op + 1, v.y * norm);
    }
}

// ---------------------------------------------------------------------------
// In-place bias (+ optional ReLU)
// ---------------------------------------------------------------------------
template <int D, bool RELU>
__global__ void bias_act_kernel(float* __restrict__ data,
                                const float* __restrict__ bias, int total) {
    int idx = blockIdx.x * blockDim.x + threadIdx.x;
    if (idx >= total) return;
    float v = data[idx] + bias[idx & (D - 1)];
    if (RELU) v = fmaxf(v, 0.0f);
    data[idx] = v;
}

// ---------------------------------------------------------------------------
// Launch
// ---------------------------------------------------------------------------
extern "C" void kernel_launch(void* const* d_in, const int* in_sizes, int n_in,
                              void* d_out, int out_size, void* d_ws, size_t ws_size,
                              hipStream_t stream) {
    const float* x    = (const float*)d_in[0];   // [N, 128]
    const int*   eidx = (const int*)  d_in[1];   // [2, E]
    const float* W1   = (const float*)d_in[2];   // [128, 128]
    const float* b1   = (const float*)d_in[3];   // [128]
    const float* W2   = (const float*)d_in[4];   // [128, 64]
    const float* b2   = (const float*)d_in[5];   // [64]
    float* out = (float*)d_out;                  // [N, 64]

    const int N = in_sizes[0] / D_IN;            // 100000 (multiple of 16)
    const int E = in_sizes[1] / 2;               // 1600000
    const int* src = eidx;
    const int* dst = eidx + E;

    // workspace layout (floats): dinv[N] | h1[N*128] | agg1[N*128]
    float* ws = (float*)d_ws;
    float* dinv = ws;
    float* h1   = ws + (((size_t)N + 255) & ~(size_t)255);
    float* agg1 = h1 + (size_t)N * D_HID;
    float* h2   = h1;                            // reuse h1 region for layer-2 GEMM out

    const int BT = 256;

    // --- normalization coefficients ---
    deg_init_kernel<<<(N + BT - 1) / BT, BT, 0, stream>>>(dinv, N);
    deg_accum_kernel<<<(E + BT - 1) / BT, BT, 0, stream>>>(dst, dinv, E);
    deg_rsqrt_kernel<<<(N + BT - 1) / BT, BT, 0, stream>>>(dinv, N);

    // --- layer 1: h1 = x @ W1 ; agg1 = sym-norm aggregate ; relu(+b1) ---
    gemm_wmma_f32_kernel<D_HID><<<N / 16, 32 * (D_HID / 16), 0, stream>>>(x, W1, h1, N);
    {
        const int total = N * D_HID;
        self_init_kernel<D_HID><<<(total + BT - 1) / BT, BT, 0, stream>>>(h1, dinv, agg1, total);
        const long tthreads = (long)E * 32;
        edge_scatter_kernel<D_HID, 4><<<(unsigned)((tthreads + BT - 1) / BT), BT, 0, stream>>>(
            src, dst, h1, dinv, agg1, E);
        bias_act_kernel<D_HID, true><<<(total + BT - 1) / BT, BT, 0, stream>>>(agg1, b1, total);
    }

    // --- layer 2: h2 = relu1 @ W2 ; d_out = sym-norm aggregate + b2 ---
    gemm_wmma_f32_kernel<D_OUT><<<N / 16, 32 * (D_OUT / 16), 0, stream>>>(agg1, W2, h2, N);
    {
        const int total = N * D_OUT;
        self_init_kernel<D_OUT><<<(total + BT - 1) / BT, BT, 0, stream>>>(h2, dinv, out, total);
        const long tthreads = (long)E * 32;
        edge_scatter_kernel<D_OUT, 2><<<(unsigned)((tthreads + BT - 1) / BT), BT, 0, stream>>>(
            src, dst, h2, dinv, out, E);
        bias_act_kernel<D_OUT, false><<<(total + BT - 1) / BT, BT, 0, stream>>>(out, b2, total);
    }
}